// LSTM_model_6545530159611
// MI455X (gfx1250) — compile-verified
//
#include <hip/hip_runtime.h>
#include <math.h>

typedef float v2f __attribute__((ext_vector_type(2)));
typedef float v8f __attribute__((ext_vector_type(8)));

#define I_SIZE 8192
#define O_SIZE 8192

// One workgroup = 8 waves = one 16-row output tile.
// Wave w computes one of the 8 matvecs for those 16 rows via V_WMMA_F32_16X16X4_F32:
//   w=0: forget_x@x   w=1: forget_h@pred
//   w=2: input_x@x    w=3: input_h@pred
//   w=4: b_input_x@x  w=5: b_input_h@pred
//   w=6: output_x@x   w=7: output_h@pred
__global__ __launch_bounds__(256) void lstm_step_kernel(
    const float* __restrict__ x,    const float* __restrict__ pred,
    const float* __restrict__ hidden,
    const float* __restrict__ Wfx,  const float* __restrict__ Wfh, const float* __restrict__ bf,
    const float* __restrict__ Wix,  const float* __restrict__ Wih, const float* __restrict__ bi,
    const float* __restrict__ Wbx,  const float* __restrict__ Wbh, const float* __restrict__ bb,
    const float* __restrict__ Wox,  const float* __restrict__ Woh, const float* __restrict__ bo,
    float* __restrict__ out)
{
    __shared__ float partial[8 * 16];   // [wave][row-in-tile]

    const int tid     = threadIdx.x;
    const int wave    = tid >> 5;       // 0..7 -> which matvec
    const int lane    = tid & 31;
    const int half    = lane >> 4;      // 0 -> K pair {0,1}, 1 -> K pair {2,3}
    const int m       = lane & 15;      // row within the 16-row tile
    const int rowbase = blockIdx.x * 16;

    // Uniform-per-wave operand selection (EXEC stays all-ones).
    const float* W;
    const float* v;
    switch (wave) {
      case 0:  W = Wfx; v = x;    break;
      case 1:  W = Wfh; v = pred; break;
      case 2:  W = Wix; v = x;    break;
      case 3:  W = Wih; v = pred; break;
      case 4:  W = Wbx; v = x;    break;
      case 5:  W = Wbh; v = pred; break;
      case 6:  W = Wox; v = x;    break;
      default: W = Woh; v = pred; break;
    }

    // A-matrix 16x4 f32 layout: VGPR0 = K{0 | 2}, VGPR1 = K{1 | 3}, lanes 0-15 = M.
    // Lane loads float2 of its row at column k + 2*half -> 8B/lane/iter, sequential 16B/row/iter,
    // full cacheline consumption across consecutive K steps.
    const float* wrow = W + (size_t)(rowbase + m) * I_SIZE + 2 * half;
    // B-matrix 4x16 f32: vector value broadcast across all 16 N columns.
    const float* vptr = v + 2 * half;

    v8f c = {};  // 16x16 f32 accumulator tile (all N columns identical)

    #pragma unroll 8
    for (int k = 0; k < I_SIZE; k += 4) {
        v2f a = *(const v2f*)(wrow + k);
        v2f b = *(const v2f*)(vptr + k);
        // D = A(16x4) * B(4x16) + C ; emits v_wmma_f32_16x16x4_f32
        c = __builtin_amdgcn_wmma_f32_16x16x4_f32(
                /*neg_a=*/false, a, /*neg_b=*/false, b,
                /*c_mod=*/(short)0, c, /*reuse_a=*/false, /*reuse_b=*/false);
    }

    // C/D layout: VGPR r holds M=r (lanes 0-15) and M=r+8 (lanes 16-31); every N
    // column equals the dot product. Lanes 0 and 16 spill the 16 row sums.
    if ((lane & 15) == 0) {
        float* dst = &partial[wave * 16 + half * 8];
        dst[0] = c[0]; dst[1] = c[1]; dst[2] = c[2]; dst[3] = c[3];
        dst[4] = c[4]; dst[5] = c[5]; dst[6] = c[6]; dst[7] = c[7];
    }
    __syncthreads();

    // Gate fusion: 16 rows handled by threads 0..15.
    if (tid < 16) {
        const int row = rowbase + tid;
        float fz = partial[0 * 16 + tid] + partial[1 * 16 + tid] + bf[row];
        float iz = partial[2 * 16 + tid] + partial[3 * 16 + tid] + bi[row];
        float bz = partial[4 * 16 + tid] + partial[5 * 16 + tid] + bb[row];
        float oz = partial[6 * 16 + tid] + partial[7 * 16 + tid] + bo[row];

        float fgate  = 1.0f / (1.0f + expf(-fz));
        float igate  = tanhf(iz);
        float bigate = 1.0f / (1.0f + expf(-bz));
        float ogate  = 1.0f / (1.0f + expf(-oz));

        float new_hidden = hidden[row] * fgate + igate * bigate;
        out[row] = tanhf(new_hidden) * ogate;
    }
}

extern "C" void kernel_launch(void* const* d_in, const int* in_sizes, int n_in,
                              void* d_out, int out_size, void* d_ws, size_t ws_size,
                              hipStream_t stream) {
    (void)in_sizes; (void)n_in; (void)out_size; (void)d_ws; (void)ws_size;

    const float* x      = (const float*)d_in[0];
    const float* pred   = (const float*)d_in[1];
    const float* hidden = (const float*)d_in[2];
    const float* Wfx    = (const float*)d_in[3];
    const float* Wfh    = (const float*)d_in[4];
    const float* bf     = (const float*)d_in[5];
    const float* Wix    = (const float*)d_in[6];
    const float* Wih    = (const float*)d_in[7];
    const float* bi     = (const float*)d_in[8];
    const float* Wbx    = (const float*)d_in[9];
    const float* Wbh    = (const float*)d_in[10];
    const float* bb     = (const float*)d_in[11];
    const float* Wox    = (const float*)d_in[12];
    const float* Woh    = (const float*)d_in[13];
    const float* bo     = (const float*)d_in[14];
    float* out          = (float*)d_out;

    dim3 grid(O_SIZE / 16);   // 512 blocks
    dim3 block(256);          // 8 waves: one per matvec
    lstm_step_kernel<<<grid, block, 0, stream>>>(
        x, pred, hidden,
        Wfx, Wfh, bf, Wix, Wih, bi, Wbx, Wbh, bb, Wox, Woh, bo,
        out);
}